// TileCoding_Joint_46402826666079
// MI455X (gfx1250) — compile-verified
//
#include <hip/hip_runtime.h>

// Tile-coding joint P/V lookup for MI455X (gfx1250).
// Memory-bound random-gather workload: 16 dword gathers/sample from 12MB of
// tables (L2-resident on a 192MB-L2 part). No dense matmul -> WMMA not
// applicable; the CDNA5 feature that helps is async global->LDS staging of the
// streamed state tile (global_load_async_to_lds_b128 + s_wait_asynccnt).

#define NBINS      256
#define NTILINGS   8
#define TABLE      (NBINS * NBINS)        // 65536 elements per (a,t) slice
#define ASTRIDE    (NTILINGS * TABLE)     // 524288 elements per action
#define BLOCK      256
#define SPT        4                      // samples per thread
#define SAMPLES_PER_BLOCK (BLOCK * SPT)   // 1024

#if __has_builtin(__builtin_amdgcn_global_load_async_to_lds_b128)
#define HAVE_ASYNC_LDS 1
#else
#define HAVE_ASYNC_LDS 0
#endif

#if HAVE_ASYNC_LDS
typedef int v4i __attribute__((ext_vector_type(4)));
typedef __attribute__((address_space(1))) v4i* g4ptr_t;   // global src
typedef __attribute__((address_space(3))) v4i* l4ptr_t;   // LDS dst
#endif

__device__ __forceinline__ float2 eval_sample(const float* __restrict__ Wp,
                                              const float* __restrict__ Wv,
                                              float p, float v,
                                              float oh1, float oh2) {
    // u in [0, 256]; per tiling t: bin = clamp(floor(u + t/8), 0, 255)
    const float INVW0 = 256.0f / 1.8f;   // 1/width for position
    const float INVW1 = 256.0f / 0.14f;  // 1/width for velocity
    float u0 = (p + 1.2f)  * INVW0;
    float u1 = (v + 0.07f) * INVW1;

    // one-hot decode: a = 0*oh0 + 1*oh1 + 2*oh2 (exact in fp32)
    unsigned a     = (unsigned)(int)fmaf(oh2, 2.0f, oh1);
    unsigned abase = a * (unsigned)ASTRIDE;

    unsigned idx[NTILINGS];
#pragma unroll
    for (int t = 0; t < NTILINGS; ++t) {
        float off = (float)t * 0.125f;
        int b0 = (int)(u0 + off);              // trunc == floor (u >= 0)
        int b1 = (int)(u1 + off);
        b0 = b0 < 0 ? 0 : (b0 > 255 ? 255 : b0);
        b1 = b1 < 0 ? 0 : (b1 > 255 ? 255 : b1);
        idx[t] = abase + ((unsigned)t << 16) + ((unsigned)b0 << 8) + (unsigned)b1;
    }

    // Issue all 16 gathers before reducing (maximize loads in flight).
    float tp[NTILINGS], tv[NTILINGS];
#pragma unroll
    for (int t = 0; t < NTILINGS; ++t) {
        tp[t] = Wp[idx[t]];
        tv[t] = Wv[idx[t]];
    }
    float sp = 0.0f, sv = 0.0f;
#pragma unroll
    for (int t = 0; t < NTILINGS; ++t) { sp += tp[t]; sv += tv[t]; }
    return make_float2(sp, sv);
}

__global__ __launch_bounds__(BLOCK)
void TileCoding_Joint_kernel(const float* __restrict__ state,
                             const float* __restrict__ Wp,
                             const float* __restrict__ Wv,
                             float* __restrict__ out,
                             int batch) {
    __shared__ float smem[SAMPLES_PER_BLOCK * 5];   // 20 KB state slab

    const int       tid  = threadIdx.x;
    const long long base = (long long)blockIdx.x * SAMPLES_PER_BLOCK;
    const bool full_block = (base + SAMPLES_PER_BLOCK) <= (long long)batch;

    if (full_block) {
        // ---- Stage this block's state slab into LDS ----
#if HAVE_ASYNC_LDS
        {
            char* g = (char*)(state + base * 5);
            char* l = (char*)smem;
#pragma unroll
            for (int k = 0; k < 5; ++k) {
                const int boff = (k * BLOCK + tid) * 16;   // dense 16B/lane
                __builtin_amdgcn_global_load_async_to_lds_b128(
                    (g4ptr_t)(g + boff), (l4ptr_t)(l + boff), 0, 0);
            }
        }
#if __has_builtin(__builtin_amdgcn_s_wait_asynccnt)
        __builtin_amdgcn_s_wait_asynccnt(0);
#else
        asm volatile("s_wait_asynccnt 0" ::: "memory");
#endif
        __syncthreads();
        const float* srow = smem + tid * (SPT * 5);
#else
        // Fallback: direct, aligned float4 loads (80B per thread, 16B aligned)
        float st[SPT * 5];
        {
            const float4* g4 = (const float4*)(state + base * 5 + (long long)tid * (SPT * 5));
            float4* s4 = (float4*)st;
#pragma unroll
            for (int k = 0; k < 5; ++k) s4[k] = g4[k];
        }
        const float* srow = st;
#endif

        // ---- Compute 4 samples per thread ----
        float2 r[SPT];
#pragma unroll
        for (int i = 0; i < SPT; ++i) {
            const float* s = srow + i * 5;
            r[i] = eval_sample(Wp, Wv, s[0], s[1], s[3], s[4]);
        }

        // ---- 32B-aligned vector store of 8 output floats ----
        float4* o4 = (float4*)(out + (base + (long long)tid * SPT) * 2);
        o4[0] = make_float4(r[0].x, r[0].y, r[1].x, r[1].y);
        o4[1] = make_float4(r[2].x, r[2].y, r[3].x, r[3].y);
    } else {
        // Tail path (never taken for BATCH = 4M, kept for generality)
#pragma unroll
        for (int i = 0; i < SPT; ++i) {
            long long s = base + (long long)tid * SPT + i;
            if (s < (long long)batch) {
                const float* sp = state + s * 5;
                float2 r = eval_sample(Wp, Wv, sp[0], sp[1], sp[3], sp[4]);
                out[s * 2 + 0] = r.x;
                out[s * 2 + 1] = r.y;
            }
        }
    }
}

extern "C" void kernel_launch(void* const* d_in, const int* in_sizes, int n_in,
                              void* d_out, int out_size, void* d_ws, size_t ws_size,
                              hipStream_t stream) {
    (void)n_in; (void)out_size; (void)d_ws; (void)ws_size;
    const float* state = (const float*)d_in[0];
    const float* Wp    = (const float*)d_in[1];
    const float* Wv    = (const float*)d_in[2];
    float*       out   = (float*)d_out;

    const int batch  = in_sizes[0] / 5;
    const int blocks = (batch + SAMPLES_PER_BLOCK - 1) / SAMPLES_PER_BLOCK;

    TileCoding_Joint_kernel<<<blocks, BLOCK, 0, stream>>>(state, Wp, Wv, out, batch);
}